// HungarianMatcherDynamicK_23235773071540
// MI455X (gfx1250) — compile-verified
//
#include <hip/hip_runtime.h>
#include <stdint.h>

// ---------------- problem constants ----------------
#define NB 16
#define NQ 4000
#define NC 80
#define NG 300
#define GPAD 304
#define FST 84          // focal LDS stride (pads 80 -> conflict-free)
#define NGT 19          // 304/16 g-tiles

typedef float v2f __attribute__((ext_vector_type(2)));
typedef float v8f __attribute__((ext_vector_type(8)));
typedef unsigned int u32x4 __attribute__((ext_vector_type(4)));
typedef int i32x8 __attribute__((ext_vector_type(8)));
typedef int i32x4 __attribute__((ext_vector_type(4)));

// ---------------- TDM: 1-D tensor load global -> LDS ----------------
// Builds a D# descriptor (groups 0/1; groups 2/3 disabled/zero) for a
// contiguous run of n 4-byte elements and issues TENSOR_LOAD_TO_LDS
// (tracked by TENSORcnt). clang-23 form: 6 args.
__device__ __forceinline__ void tdm_load_1d(const void* gsrc, void* ldst, unsigned n) {
#if __has_builtin(__builtin_amdgcn_tensor_load_to_lds)
  unsigned long long ga = (unsigned long long)(uintptr_t)gsrc;
  unsigned lds = (unsigned)(uintptr_t)ldst;          // low 32 bits of flat = LDS offset
  u32x4 g0;
  g0[0] = 1u;                                        // count=1, user mode
  g0[1] = lds;                                       // lds_addr
  g0[2] = (unsigned)(ga & 0xFFFFFFFFull);            // global_addr[31:0]
  g0[3] = (unsigned)((ga >> 32) & 0x01FFFFFFull) | (2u << 30); // addr[56:32] | type=2
  i32x8 g1;
  g1[0] = (int)(2u << 16);                           // workgroup_mask=0, data_size=4B
  g1[1] = (int)((n & 0xFFFFu) << 16);                // tensor_dim0[15:0]
  g1[2] = (int)((n >> 16) | (1u << 16));             // tensor_dim0[31:16], tensor_dim1=1
  g1[3] = (int)((n & 0xFFFFu) << 16);                // tile_dim0 = n
  g1[4] = 1;                                         // tile_dim1=1, tile_dim2=0
  g1[5] = (int)n;                                    // tensor_dim0_stride lo
  g1[6] = 0;
  g1[7] = 0;
  i32x4 z4 = {0, 0, 0, 0};
  i32x8 z8 = {0, 0, 0, 0, 0, 0, 0, 0};
  __builtin_amdgcn_tensor_load_to_lds(g0, g1, z4, z4, z8, 0);
#else
  (void)gsrc; (void)ldst; (void)n;
#endif
}

__device__ __forceinline__ void tdm_wait_all() {
#if __has_builtin(__builtin_amdgcn_s_wait_tensorcnt)
  __builtin_amdgcn_s_wait_tensorcnt(0);
#endif
}

__device__ __forceinline__ float focal_term(float lg, float conf) {
  float s = 1.0f / (1.0f + __expf(-lg));
  float p = sqrtf(s * conf);
  float pos = 0.25f * (1.0f - p) * (1.0f - p) * (-__logf(p + 1e-8f));
  float neg = 0.75f * p * p * (-__logf(1.0f - p + 1e-8f));
  return pos - neg;
}

__device__ __forceinline__ unsigned fkey(float f) {
  unsigned u = __float_as_uint(f);
  return (u & 0x80000000u) ? ~u : (u | 0x80000000u);
}

// =====================================================================
// Phase A: cost + iou matrices. grid(ceil(Q/64), B), block 128 (4 waves).
// Each wave owns a 16-query strip; class cost via fp32 WMMA one-hot GEMM.
// =====================================================================
__global__ __launch_bounds__(128) void cost_kernel(
    const float* __restrict__ logits, const float* __restrict__ scores,
    const float* __restrict__ boxes, const float* __restrict__ gts,
    const int* __restrict__ labels, const float* __restrict__ imgsz,
    float* __restrict__ outCost, float* __restrict__ outIou) {
  __shared__ float sFocP[4][16][FST];
  __shared__ float sFocC[4][16][FST];
  __shared__ float sGtP[GPAD][4];
  __shared__ float sGtC[GPAD][4];
  __shared__ float sGAreaP[GPAD];
  __shared__ float sGAreaC[GPAD];
  __shared__ int   sLabP[GPAD];
  __shared__ int   sLabC[GPAD];
  __shared__ float sBoxP[64][4];
  __shared__ float sBoxC[64][4];
  __shared__ float sConf[64];
  __shared__ float sCxP[64], sCyP[64], sCxC[64], sCyC[64];
  __shared__ float sAQP[64], sAQC[64];
  __shared__ float sInvImg[8];

  const int tid  = threadIdx.x;
  const int lane = tid & 31;
  const int w    = tid >> 5;
  const int half = lane >> 4;
  const int ln   = lane & 15;
  const int bat  = blockIdx.y;
  const int qBlock = blockIdx.x * 64;

  // --- wave 0: kick off GT staging via the Tensor Data Mover ---
  if (w == 0) {
    tdm_load_1d(gts + (size_t)bat * NG * 4,        &sGtP[0][0], NG * 4);
    tdm_load_1d(gts + (size_t)(NB + bat) * NG * 4, &sGtC[0][0], NG * 4);
    tdm_load_1d(labels + (size_t)bat * NG,         &sLabP[0],   NG);
    tdm_load_1d(labels + (size_t)(NB + bat) * NG,  &sLabC[0],   NG);
  }

  // --- per-query staging (disjoint LDS from TDM targets) ---
  for (int i = tid; i < 64; i += 128) {
    int q = qBlock + i;
    bool v = q < NQ;
    sConf[i] = v ? scores[(size_t)bat * NQ + q] : 0.5f;
    float p0 = 0.f, p1 = 0.f, p2 = 1.f, p3 = 1.f;
    float c0 = 0.f, c1 = 0.f, c2 = 1.f, c3 = 1.f;
    if (v) {
      const float* bp = boxes + ((size_t)bat * NQ + q) * 4;
      const float* bc = boxes + ((size_t)(NB + bat) * NQ + q) * 4;
      p0 = bp[0]; p1 = bp[1]; p2 = bp[2]; p3 = bp[3];
      c0 = bc[0]; c1 = bc[1]; c2 = bc[2]; c3 = bc[3];
    }
    sBoxP[i][0] = p0; sBoxP[i][1] = p1; sBoxP[i][2] = p2; sBoxP[i][3] = p3;
    sBoxC[i][0] = c0; sBoxC[i][1] = c1; sBoxC[i][2] = c2; sBoxC[i][3] = c3;
    sCxP[i] = 0.5f * (p0 + p2); sCyP[i] = 0.5f * (p1 + p3);
    sCxC[i] = 0.5f * (c0 + c2); sCyC[i] = 0.5f * (c1 + c3);
    sAQP[i] = (p2 - p0) * (p3 - p1);
    sAQC[i] = (c2 - c0) * (c3 - c1);
  }
  if (tid < 4) {
    sInvImg[tid]     = 1.0f / imgsz[(size_t)bat * 4 + tid];
    sInvImg[4 + tid] = 1.0f / imgsz[(size_t)(NB + bat) * 4 + tid];
    int g = NG + tid;  // pad entries (g-tile padding)
    sLabP[g] = -1; sLabC[g] = -1;
    sGtP[g][0] = 0.f; sGtP[g][1] = 0.f; sGtP[g][2] = 1.f; sGtP[g][3] = 1.f;
    sGtC[g][0] = 0.f; sGtC[g][1] = 0.f; sGtC[g][2] = 1.f; sGtC[g][3] = 1.f;
    sGAreaP[g] = 1.f; sGAreaC[g] = 1.f;
  }
  __syncthreads();

  // --- focal tables (overlaps TDM DMA) ---
  for (int t = lane; t < 16 * NC; t += 32) {
    int m = t / NC;
    int c = t - m * NC;
    int i = w * 16 + m;
    int q = qBlock + i;
    float cf = sConf[i];
    float lgP = 0.f, lgC = 0.f;
    if (q < NQ) {
      lgP = logits[((size_t)bat * NQ + q) * NC + c];
      lgC = logits[((size_t)(NB + bat) * NQ + q) * NC + c];
    }
    sFocP[w][m][c] = focal_term(lgP, cf);
    sFocC[w][m][c] = focal_term(lgC, cf);
  }

  if (w == 0) tdm_wait_all();
  __syncthreads();

  // --- gt areas ---
  for (int t = tid; t < 2 * NG; t += 128) {
    if (t < NG) sGAreaP[t] = (sGtP[t][2] - sGtP[t][0]) * (sGtP[t][3] - sGtP[t][1]);
    else { int g = t - NG; sGAreaC[g] = (sGtC[g][2] - sGtC[g][0]) * (sGtC[g][3] - sGtC[g][1]); }
  }
  __syncthreads();

  // --- pass 1: fg[q] = (any in_box | any in_ctr) pre & cur, via wave32 ballots ---
  const int iQ = w * 16 + ln;
  float axP = sCxP[iQ], ayP = sCyP[iQ], axC = sCxC[iQ], ayC = sCyC[iQ];
  bool anyBoxP = false, anyCtrP = false, anyBoxC = false, anyCtrC = false;
  for (int g = half; g < NG; g += 2) {
    float x1 = sGtP[g][0], y1 = sGtP[g][1], x2 = sGtP[g][2], y2 = sGtP[g][3];
    anyBoxP = anyBoxP || ((axP > x1) && (axP < x2) && (ayP > y1) && (ayP < y2));
    float cx = 0.5f * (x1 + x2), cy = 0.5f * (y1 + y2);
    float ww = 2.5f * (x2 - x1), hh = 2.5f * (y2 - y1);
    anyCtrP = anyCtrP || ((axP > cx - ww) && (axP < cx + ww) && (ayP > cy - hh) && (ayP < cy + hh));
    x1 = sGtC[g][0]; y1 = sGtC[g][1]; x2 = sGtC[g][2]; y2 = sGtC[g][3];
    anyBoxC = anyBoxC || ((axC > x1) && (axC < x2) && (ayC > y1) && (ayC < y2));
    cx = 0.5f * (x1 + x2); cy = 0.5f * (y1 + y2);
    ww = 2.5f * (x2 - x1); hh = 2.5f * (y2 - y1);
    anyCtrC = anyCtrC || ((axC > cx - ww) && (axC < cx + ww) && (ayC > cy - hh) && (ayC < cy + hh));
  }
  unsigned mBoxP = (unsigned)__ballot(anyBoxP);
  unsigned mCtrP = (unsigned)__ballot(anyCtrP);
  unsigned mBoxC = (unsigned)__ballot(anyBoxC);
  unsigned mCtrC = (unsigned)__ballot(anyCtrC);
  unsigned aPm = (mBoxP | (mBoxP >> 16) | mCtrP | (mCtrP >> 16)) & 0xFFFFu;
  unsigned aCm = (mBoxC | (mBoxC >> 16) | mCtrC | (mCtrC >> 16)) & 0xFFFFu;
  unsigned fgMask = aPm & aCm;

  // --- pass 2: 16x16 (q,g) tiles; class cost on the matrix pipe ---
  float i0 = sInvImg[0], i1 = sInvImg[1], i2 = sInvImg[2], i3 = sInvImg[3];
  float j0 = sInvImg[4], j1 = sInvImg[5], j2 = sInvImg[6], j3 = sInvImg[7];

  for (int gt = 0; gt < NGT; ++gt) {
    int gg = gt * 16 + ln;
    float gpx1 = sGtP[gg][0], gpy1 = sGtP[gg][1], gpx2 = sGtP[gg][2], gpy2 = sGtP[gg][3];
    float gcx1 = sGtC[gg][0], gcy1 = sGtC[gg][1], gcx2 = sGtC[gg][2], gcy2 = sGtC[gg][3];
    float gAP = sGAreaP[gg], gAC = sGAreaC[gg];
    int labP = sLabP[gg], labC = sLabC[gg];
    float pcx = 0.5f * (gpx1 + gpx2), pcy = 0.5f * (gpy1 + gpy2);
    float pw = 2.5f * (gpx2 - gpx1), ph = 2.5f * (gpy2 - gpy1);
    float pLx = pcx - pw, pHx = pcx + pw, pLy = pcy - ph, pHy = pcy + ph;
    float ccx = 0.5f * (gcx1 + gcx2), ccy = 0.5f * (gcy1 + gcy2);
    float cw = 2.5f * (gcx2 - gcx1), ch = 2.5f * (gcy2 - gcy1);
    float cLx = ccx - cw, cHx = ccx + cw, cLy = ccy - ch, cHy = ccy + ch;

#if __has_builtin(__builtin_amdgcn_wmma_f32_16x16x4_f32)
    v8f acc = {0.f, 0.f, 0.f, 0.f, 0.f, 0.f, 0.f, 0.f};
#pragma unroll
    for (int kb = 0; kb < NC; kb += 4) {
      int kk = kb + 2 * half;
      v2f A  = { sFocP[w][ln][kk], sFocP[w][ln][kk + 1] };
      v2f Bm = { (labP == kk) ? 1.f : 0.f, (labP == kk + 1) ? 1.f : 0.f };
      acc = __builtin_amdgcn_wmma_f32_16x16x4_f32(false, A, false, Bm, (short)0, acc, false, false);
      v2f A2 = { sFocC[w][ln][kk], sFocC[w][ln][kk + 1] };
      v2f B2 = { (labC == kk) ? 1.f : 0.f, (labC == kk + 1) ? 1.f : 0.f };
      acc = __builtin_amdgcn_wmma_f32_16x16x4_f32(false, A2, false, B2, (short)0, acc, false, false);
    }
#endif

#pragma unroll
    for (int r = 0; r < 8; ++r) {
      int qL = r + 8 * half;
      int i = w * 16 + qL;
      int q = qBlock + i;
      float b0 = sBoxP[i][0], b1 = sBoxP[i][1], b2 = sBoxP[i][2], b3 = sBoxP[i][3];
      float d0 = sBoxC[i][0], d1 = sBoxC[i][1], d2 = sBoxC[i][2], d3 = sBoxC[i][3];
      // pre frame
      float cbP = fabsf(b0 - gpx1) * i0 + fabsf(b1 - gpy1) * i1 +
                  fabsf(b2 - gpx2) * i2 + fabsf(b3 - gpy2) * i3;
      float interP = fmaxf(fminf(b2, gpx2) - fmaxf(b0, gpx1), 0.f) *
                     fmaxf(fminf(b3, gpy2) - fmaxf(b1, gpy1), 0.f);
      float uniP = sAQP[i] + gAP - interP;
      float iouP = interP / uniP;
      float encP = (fmaxf(b2, gpx2) - fminf(b0, gpx1)) * (fmaxf(b3, gpy2) - fminf(b1, gpy1));
      float giouP = iouP - (encP - uniP) / encP;
      float axp = sCxP[i], ayp = sCyP[i];
      bool inbP = (axp > gpx1) && (axp < gpx2) && (ayp > gpy1) && (ayp < gpy2);
      bool ictP = (axp > pLx) && (axp < pHx) && (ayp > pLy) && (ayp < pHy);
      // cur frame
      float cbC = fabsf(d0 - gcx1) * j0 + fabsf(d1 - gcy1) * j1 +
                  fabsf(d2 - gcx2) * j2 + fabsf(d3 - gcy2) * j3;
      float interC = fmaxf(fminf(d2, gcx2) - fmaxf(d0, gcx1), 0.f) *
                     fmaxf(fminf(d3, gcy2) - fmaxf(d1, gcy1), 0.f);
      float uniC = sAQC[i] + gAC - interC;
      float iouC = interC / uniC;
      float encC = (fmaxf(d2, gcx2) - fminf(d0, gcx1)) * (fmaxf(d3, gcy2) - fminf(d1, gcy1));
      float giouC = iouC - (encC - uniC) / encC;
      float axc = sCxC[i], ayc = sCyC[i];
      bool inbCb = (axc > gcx1) && (axc < gcx2) && (ayc > gcy1) && (ayc < gcy2);
      bool ictC = (axc > cLx) && (axc < cHx) && (ayc > cLy) && (ayc < cHy);

#if __has_builtin(__builtin_amdgcn_wmma_f32_16x16x4_f32)
      float clsCost = acc[r];
#else
      float clsCost = (labP >= 0 ? sFocP[w][qL][labP] : 0.f) +
                      (labC >= 0 ? sFocC[w][qL][labC] : 0.f);
#endif
      bool inbc = inbP && ictP && inbCb && ictC;
      bool fgq = (fgMask >> qL) & 1u;
      float cost = 2.5f * (cbP + cbC) + clsCost - (giouP + giouC) +
                   (inbc ? 0.f : 100.f) + (fgq ? 0.f : 10000.f);
      float iou = 0.5f * (iouP + iouC);
      if (q < NQ && gg < NG) {
        size_t off = ((size_t)bat * NQ + q) * (size_t)NG + gg;
        outCost[off] = cost;
        outIou[off] = iou;
      }
    }
  }
}

// =====================================================================
// Phase B: per (b,g) column: top-5 ious -> dyn_k; 5 smallest costs -> colSel
// grid(G, B), block 128
// =====================================================================
__global__ __launch_bounds__(128) void col_topk_kernel(
    const float* __restrict__ ious, const float* __restrict__ cost,
    int* __restrict__ dynK, int* __restrict__ colSel) {
  __shared__ float shIv[128 * 5];
  __shared__ float shCv[128 * 5];
  __shared__ int   shCi[128 * 5];
  const int g = blockIdx.x, b = blockIdx.y, tid = threadIdx.x;
  const float* iouCol = ious + ((size_t)b * NQ) * NG + g;
  const float* cCol = cost + ((size_t)b * NQ) * NG + g;
  float tv[5], cv[5]; int ci[5];
#pragma unroll
  for (int j = 0; j < 5; ++j) { tv[j] = -1e30f; cv[j] = 1e30f; ci[j] = 0x7fffffff; }
  for (int q = tid; q < NQ; q += 128) {
    float u = iouCol[(size_t)q * NG];
    if (u > tv[4]) {
      tv[4] = u;
#pragma unroll
      for (int j = 4; j > 0; --j)
        if (tv[j] > tv[j - 1]) { float t = tv[j]; tv[j] = tv[j - 1]; tv[j - 1] = t; }
    }
    float c = cCol[(size_t)q * NG];
    if (c < cv[4] || (c == cv[4] && q < ci[4])) {
      cv[4] = c; ci[4] = q;
#pragma unroll
      for (int j = 4; j > 0; --j) {
        bool sw = (cv[j] < cv[j - 1]) || (cv[j] == cv[j - 1] && ci[j] < ci[j - 1]);
        if (sw) {
          float t = cv[j]; cv[j] = cv[j - 1]; cv[j - 1] = t;
          int ti = ci[j]; ci[j] = ci[j - 1]; ci[j - 1] = ti;
        }
      }
    }
  }
#pragma unroll
  for (int j = 0; j < 5; ++j) { shIv[tid * 5 + j] = tv[j]; shCv[tid * 5 + j] = cv[j]; shCi[tid * 5 + j] = ci[j]; }
  __syncthreads();
  for (int s = 64; s >= 1; s >>= 1) {
    if (tid < s) {
      int o = (tid + s) * 5;
#pragma unroll
      for (int j = 0; j < 5; ++j) {
        float u = shIv[o + j];
        if (u > tv[4]) {
          tv[4] = u;
#pragma unroll
          for (int k = 4; k > 0; --k)
            if (tv[k] > tv[k - 1]) { float t = tv[k]; tv[k] = tv[k - 1]; tv[k - 1] = t; }
        }
        float c = shCv[o + j]; int qi = shCi[o + j];
        if (c < cv[4] || (c == cv[4] && qi < ci[4])) {
          cv[4] = c; ci[4] = qi;
#pragma unroll
          for (int k = 4; k > 0; --k) {
            bool sw = (cv[k] < cv[k - 1]) || (cv[k] == cv[k - 1] && ci[k] < ci[k - 1]);
            if (sw) {
              float t = cv[k]; cv[k] = cv[k - 1]; cv[k - 1] = t;
              int ti = ci[k]; ci[k] = ci[k - 1]; ci[k - 1] = ti;
            }
          }
        }
      }
#pragma unroll
      for (int j = 0; j < 5; ++j) { shIv[tid * 5 + j] = tv[j]; shCv[tid * 5 + j] = cv[j]; shCi[tid * 5 + j] = ci[j]; }
    }
    __syncthreads();
  }
  if (tid == 0) {
    float ssum = tv[0] + tv[1] + tv[2] + tv[3] + tv[4];
    int k = (int)ssum;
    if (k < 1) k = 1;
    if (k > 5) k = 5;
    dynK[(size_t)b * NG + g] = k;
#pragma unroll
    for (int j = 0; j < 5; ++j) colSel[((size_t)b * NG + g) * 5 + j] = ci[j];
  }
}

// =====================================================================
// Phase B2: per-row argmin of cost (first-min). grid(ceil(Q/256), B)
// =====================================================================
__global__ __launch_bounds__(256) void rowargmin_kernel(
    const float* __restrict__ cost, int* __restrict__ rowArg) {
  int q = blockIdx.x * 256 + threadIdx.x;
  int b = blockIdx.y;
  if (q >= NQ) return;
  const float* row = cost + ((size_t)b * NQ + q) * NG;
  float bv = row[0]; int bg = 0;
  for (int g = 1; g < NG; ++g) {
    float v = row[g];
    if (v < bv) { bv = v; bg = g; }
  }
  rowArg[(size_t)b * NQ + q] = bg;
}

// =====================================================================
// Phase C: exact dynamic-k while-loop, one block per batch, state in LDS.
// =====================================================================
__global__ __launch_bounds__(1024) void assign_kernel(
    const float* __restrict__ cost, const int* __restrict__ dynKArr,
    const int* __restrict__ colSel, const int* __restrict__ rowArg,
    float* __restrict__ outSel, float* __restrict__ outGti, float* __restrict__ outMq) {
  __shared__ int rowCol[NQ];
  __shared__ float rowPen[NQ];
  __shared__ int rowAdd[NQ];
  __shared__ int colCnt[NG];
  __shared__ int pick[NG];
  __shared__ int unmL[NG];
  __shared__ int numUnm;
  __shared__ unsigned long long colBest[NG];

  const int b = blockIdx.x, tid = threadIdx.x;
  const int lane = tid & 31, wid = tid >> 5;
  const float* cb = cost + (size_t)b * NQ * NG;
  const int* selB = colSel + (size_t)b * NG * 5;
  const int* argB = rowArg + (size_t)b * NQ;
  const int* dkB = dynKArr + (size_t)b * NG;

  for (int q = tid; q < NQ; q += 1024) { rowCol[q] = -1; rowPen[q] = 0.f; rowAdd[q] = 0; }
  for (int g = tid; g < NG; g += 1024) colCnt[g] = 0;
  __syncthreads();

  // initial m: dyn_k smallest per column, then row conflicts -> row argmin
  for (int t = tid; t < NG * 5; t += 1024) {
    int g = t / 5, j = t - g * 5;
    if (j < dkB[g]) atomicAdd(&rowAdd[selB[g * 5 + j]], 1);
  }
  __syncthreads();
  for (int t = tid; t < NG * 5; t += 1024) {
    int g = t / 5, j = t - g * 5;
    if (j < dkB[g]) {
      int q = selB[g * 5 + j];
      if (rowAdd[q] == 1) { rowCol[q] = g; atomicAdd(&colCnt[g], 1); }
    }
  }
  __syncthreads();
  for (int q = tid; q < NQ; q += 1024) {
    if (rowAdd[q] > 1) { int gs = argB[q]; rowCol[q] = gs; atomicAdd(&colCnt[gs], 1); }
  }
  __syncthreads();

  for (int it = 0; it < 4096; ++it) {
    if (tid == 0) numUnm = 0;
    __syncthreads();
    for (int g = tid; g < NG; g += 1024)
      if (colCnt[g] == 0) { int p = atomicAdd(&numUnm, 1); unmL[p] = g; }
    __syncthreads();
    int nu = numUnm;
    if (nu == 0) break;
    // c2 = c + 1e5 * (row assigned); penalties accumulate across iterations
    for (int q = tid; q < NQ; q += 1024) {
      rowAdd[q] = 0;
      if (rowCol[q] >= 0) rowPen[q] += 100000.0f;
    }
    __syncthreads();
    // each unmatched column: argmin_q (cost + pen), first-min tie rule
    for (int u = wid; u < nu; u += 32) {
      int g = unmL[u];
      float bv = 3.4e38f; int bq = 0x7fffffff;
      for (int q = lane; q < NQ; q += 32) {
        float v = cb[(size_t)q * NG + g] + rowPen[q];
        if (v < bv || (v == bv && q < bq)) { bv = v; bq = q; }
      }
      for (int off = 16; off > 0; off >>= 1) {
        float ov = __shfl_xor(bv, off, 32);
        int oq = __shfl_xor(bq, off, 32);
        if (ov < bv || (ov == bv && oq < bq)) { bv = ov; bq = oq; }
      }
      if (lane == 0) { pick[g] = bq; atomicAdd(&rowAdd[bq], 1); }
    }
    __syncthreads();
    // rows with >1 assignment -> one-hot at row argmin (argmin of c2 row == argmin of cost row)
    for (int q = tid; q < NQ; q += 1024) {
      int oldg = rowCol[q];
      int total = (oldg >= 0 ? 1 : 0) + rowAdd[q];
      if (total > 1) {
        if (oldg >= 0) atomicSub(&colCnt[oldg], 1);
        int gs = argB[q];
        rowCol[q] = gs;
        atomicAdd(&colCnt[gs], 1);
      }
    }
    __syncthreads();
    // singleton new picks
    for (int u = tid; u < nu; u += 1024) {
      int g = unmL[u];
      int q = pick[g];
      if (rowCol[q] == -1 && rowAdd[q] == 1) { rowCol[q] = g; atomicAdd(&colCnt[g], 1); }
    }
    __syncthreads();
  }

  // outputs
  for (int g = tid; g < NG; g += 1024) colBest[g] = ~0ull;
  __syncthreads();
  for (int q = tid; q < NQ; q += 1024) {
    int g = rowCol[q];
    outSel[(size_t)b * NQ + q] = (g >= 0) ? 1.0f : 0.0f;
    outGti[(size_t)b * NQ + q] = (g >= 0) ? (float)g : 0.0f;
    if (g >= 0) {
      float c2 = cb[(size_t)q * NG + g] + rowPen[q];
      unsigned long long key = ((unsigned long long)fkey(c2) << 32) | (unsigned)q;
      atomicMin(&colBest[g], key);
    }
  }
  __syncthreads();
  for (int g = tid; g < NG; g += 1024) {
    unsigned long long kk = colBest[g];
    outMq[(size_t)b * NG + g] = (kk == ~0ull) ? 0.0f : (float)(unsigned)(kk & 0xffffffffu);
  }
}

// =====================================================================
extern "C" void kernel_launch(void* const* d_in, const int* in_sizes, int n_in,
                              void* d_out, int out_size, void* d_ws, size_t ws_size,
                              hipStream_t stream) {
  (void)in_sizes; (void)n_in; (void)out_size; (void)ws_size;
  const float* logits = (const float*)d_in[0];
  const float* scores = (const float*)d_in[1];
  const float* boxes  = (const float*)d_in[2];
  const float* gts    = (const float*)d_in[3];
  const int*   labels = (const int*)d_in[4];
  const float* imgsz  = (const float*)d_in[5];

  float* out = (float*)d_out;
  float* outSel  = out;
  float* outGti  = out + (size_t)NB * NQ;
  float* outMq   = out + (size_t)2 * NB * NQ;
  float* outCost = out + (size_t)2 * NB * NQ + (size_t)NB * NG;

  char* wsb = (char*)d_ws;
  float* wsIous  = (float*)wsb;
  int* wsDynK    = (int*)(wsb + (size_t)NB * NQ * NG * sizeof(float));
  int* wsColSel  = wsDynK + (size_t)NB * NG;
  int* wsRowArg  = wsColSel + (size_t)NB * NG * 5;

  dim3 gA((NQ + 63) / 64, NB);
  cost_kernel<<<gA, 128, 0, stream>>>(logits, scores, boxes, gts, labels, imgsz,
                                      outCost, wsIous);
  dim3 gB(NG, NB);
  col_topk_kernel<<<gB, 128, 0, stream>>>(wsIous, outCost, wsDynK, wsColSel);
  dim3 gB2((NQ + 255) / 256, NB);
  rowargmin_kernel<<<gB2, 256, 0, stream>>>(outCost, wsRowArg);
  assign_kernel<<<NB, 1024, 0, stream>>>(outCost, wsDynK, wsColSel, wsRowArg,
                                         outSel, outGti, outMq);
}